// CoLoc_Sel_41274635714794
// MI455X (gfx1250) — compile-verified
//
#include <hip/hip_runtime.h>
#include <hip/hip_bf16.h>
#include <math.h>

typedef __attribute__((ext_vector_type(2))) float v2f;
typedef __attribute__((ext_vector_type(8))) float v8f;

#define EPSV 1e-8f
#define B_ 4
#define D_ 512
#define Dc_ 256
#define PLANE 65536      // 256*256
#define HW_ 196          // 14*14
#define OUT_C 1024

// ---------------------------------------------------------------------------
// Kernel 1: per (b,d) plane. Copy x -> out[:, 512+d, :, :] (B128 vectorized)
// fused with plane-max reduction -> ws xp[b*512+d]. Reads x exactly once.
// ---------------------------------------------------------------------------
__global__ __launch_bounds__(256)
void k_reduce_copy(const float* __restrict__ x, float* __restrict__ out,
                   float* __restrict__ xp) {
  int plane = blockIdx.x;          // b*512 + d
  int b = plane >> 9;
  int d = plane & 511;
  const float4* src = (const float4*)(x + (size_t)plane * PLANE);
  float4* dst = (float4*)(out + ((size_t)b * OUT_C + 512 + d) * PLANE);
  float m = -INFINITY;
  for (int i = threadIdx.x; i < PLANE / 4; i += 256) {
    float4 v = src[i];
    dst[i] = v;
    m = fmaxf(m, fmaxf(fmaxf(v.x, v.y), fmaxf(v.z, v.w)));
  }
  for (int off = 16; off > 0; off >>= 1)
    m = fmaxf(m, __shfl_xor(m, off, 32));
  __shared__ float red[8];
  int lane = threadIdx.x & 31, wid = threadIdx.x >> 5;
  if (lane == 0) red[wid] = m;
  __syncthreads();
  if (threadIdx.x == 0) {
    float r = red[0];
    for (int i = 1; i < 8; ++i) r = fmaxf(r, red[i]);
    xp[plane] = r;
  }
}

// ---------------------------------------------------------------------------
// Kernel 2: matching computation. 1 block, 8 waves; wave (b,c) computes the
// (p=2) x (hw=196) similarity map as a WMMA matmul over K=256, then the block
// cooperates through LDS for scores / winner / loss / att_maps / v_sel.
// ---------------------------------------------------------------------------
__global__ __launch_bounds__(256)
void k_match(const float* __restrict__ v0, const float* __restrict__ v1,
             const float* __restrict__ xp_g, float* __restrict__ out,
             float* __restrict__ vsel) {
  __shared__ float xp_l[B_][D_];           // 8 KB
  __shared__ float vn_l[B_][2][HW_];       // 6.1 KB
  __shared__ float maps_l[B_][2][2][HW_];  // 12.3 KB
  __shared__ float xn_l[B_][2];
  __shared__ float smax_l[B_][2][2];
  __shared__ float loss_l[B_];

  const int tid = threadIdx.x;
  const int lane = tid & 31;
  const int wid = tid >> 5;   // 0..7
  const int b = wid >> 1;
  const int c = wid & 1;

  for (int i = tid; i < B_ * D_; i += 256) ((float*)xp_l)[i] = xp_g[i];
  __syncthreads();

  const float* row0 = &xp_l[b][c * Dc_];        // x_t[b, p=0, c, :]
  const float* row1 = &xp_l[b][(1 - c) * Dc_];  // x_t[b, p=1, c, :]

  // xn0[b][c] = ||xp[b,c,:]||
  float s = 0.f;
  for (int d = lane; d < Dc_; d += 32) { float t = row0[d]; s += t * t; }
  for (int off = 16; off > 0; off >>= 1) s += __shfl_xor(s, off, 32);
  if (lane == 0) xn_l[b][c] = sqrtf(s);

  const float* vbase = (c == 0 ? v0 : v1) + (size_t)b * Dc_ * HW_;

  // vn[b][c][hw] = ||v_cat[b,c,:,hw]||
  for (int hw = lane; hw < HW_; hw += 32) {
    float acc = 0.f;
    for (int d = 0; d < Dc_; ++d) {
      float t = vbase[(size_t)d * HW_ + hw];
      acc += t * t;
    }
    vn_l[b][c][hw] = sqrtf(acc);
  }
  __syncthreads();

  const int m = lane & 15;
  const int khalf = (lane >> 4) << 1;  // lanes 16-31 hold K=k0+2,k0+3

  // dot[p, n] = sum_d A[p,d] * v[d,n] via v_wmma_f32_16x16x4_f32
  for (int nt = 0; nt < 13; ++nt) {
    int n = nt * 16 + m;
    int hw = n < HW_ ? n : HW_ - 1;
    v8f acc = {0.f, 0.f, 0.f, 0.f, 0.f, 0.f, 0.f, 0.f};
    for (int k0 = 0; k0 < Dc_; k0 += 4) {
      int ka = k0 + khalf;
      v2f a, bb;
      // A: M=0 -> p=0 row, M=1 -> p=1 row, other rows zero
      a.x = (m == 0) ? row0[ka]     : ((m == 1) ? row1[ka]     : 0.f);
      a.y = (m == 0) ? row0[ka + 1] : ((m == 1) ? row1[ka + 1] : 0.f);
      float bx = vbase[(size_t)ka * HW_ + hw];
      float by = vbase[(size_t)(ka + 1) * HW_ + hw];
      bb.x = (n < HW_) ? bx : 0.f;
      bb.y = (n < HW_) ? by : 0.f;
      acc = __builtin_amdgcn_wmma_f32_16x16x4_f32(
          false, a, false, bb, (short)0, acc, false, false);
    }
    // D layout: lanes 0-15, VGPR r -> M=r, N=lane
    if (lane < 16 && n < HW_) {
      float vnv = vn_l[b][c][n];
      maps_l[b][c][0][n] = acc[0] / fmaxf(xn_l[b][c] * vnv, EPSV);
      maps_l[b][c][1][n] = acc[1] / fmaxf(xn_l[b][1 - c] * vnv, EPSV);
    }
  }

  // per-(b,c,p) spatial max
  for (int p = 0; p < 2; ++p) {
    float mm = -INFINITY;
    for (int hw = lane; hw < HW_; hw += 32) mm = fmaxf(mm, maps_l[b][c][p][hw]);
    for (int off = 16; off > 0; off >>= 1)
      mm = fmaxf(mm, __shfl_xor(mm, off, 32));
    if (lane == 0) smax_l[b][c][p] = mm;
  }
  __syncthreads();

  float s0 = smax_l[b][0][0] + smax_l[b][1][0];
  float s1 = smax_l[b][0][1] + smax_l[b][1][1];
  int win = (s1 > s0) ? 1 : 0;             // stable argsort: ties keep p=0
  float lossb = win ? (s0 - s1) : (s1 - s0);  // ss[1] - ss[0]

  // att_maps out + argmax (first max index) over hw
  float bv = -INFINITY;
  int bi = 0;
  size_t att_off = (size_t)B_ * OUT_C * PLANE + 1 + (size_t)(b * 2 + c) * HW_;
  for (int hw = lane; hw < HW_; hw += 32) {
    float mv = maps_l[b][c][win][hw];
    out[att_off + hw] = mv;
    if (mv > bv) { bv = mv; bi = hw; }
  }
  for (int off = 16; off > 0; off >>= 1) {
    float ov = __shfl_xor(bv, off, 32);
    int oi = __shfl_xor(bi, off, 32);
    if (ov > bv || (ov == bv && oi < bi)) { bv = ov; bi = oi; }
  }

  // v_sel[b,c,:] = v_cat[b,c,:,argmax]
  for (int d = lane; d < Dc_; d += 32)
    vsel[(size_t)(b * 2 + c) * Dc_ + d] = vbase[(size_t)d * HW_ + bi];

  if (c == 0 && lane == 0) loss_l[b] = lossb;
  __syncthreads();
  if (tid == 0)
    out[(size_t)B_ * OUT_C * PLANE] =
        0.25f * (loss_l[0] + loss_l[1] + loss_l[2] + loss_l[3]);
}

// ---------------------------------------------------------------------------
// Kernel 3: broadcast v_sel into out[:, 0:512, :, :] with B128 stores.
// ---------------------------------------------------------------------------
__global__ __launch_bounds__(256)
void k_broadcast(const float* __restrict__ vsel, float* __restrict__ out) {
  int blk = blockIdx.x;  // b*512 + ch, ch = c*256 + d
  int b = blk >> 9;
  int ch = blk & 511;
  float val = vsel[b * 512 + ch];
  float4 v4 = make_float4(val, val, val, val);
  float4* dst = (float4*)(out + ((size_t)b * OUT_C + ch) * PLANE);
  for (int i = threadIdx.x; i < PLANE / 4; i += 256) dst[i] = v4;
}

// ---------------------------------------------------------------------------
extern "C" void kernel_launch(void* const* d_in, const int* in_sizes, int n_in,
                              void* d_out, int out_size, void* d_ws,
                              size_t ws_size, hipStream_t stream) {
  const float* x  = (const float*)d_in[0];
  const float* v0 = (const float*)d_in[1];
  const float* v1 = (const float*)d_in[2];
  float* out = (float*)d_out;
  float* ws = (float*)d_ws;
  float* xp   = ws;         // 2048 floats: per-channel plane maxima
  float* vsel = ws + 2048;  // 2048 floats: selected v vectors

  k_reduce_copy<<<B_ * D_, 256, 0, stream>>>(x, out, xp);
  k_match<<<1, 256, 0, stream>>>(v0, v1, xp, out, vsel);
  k_broadcast<<<B_ * 512, 256, 0, stream>>>(vsel, out);
}